// KmeansAttention_23553600651620
// MI455X (gfx1250) — compile-verified
//
#include <hip/hip_runtime.h>
#include <hip/hip_bf16.h>

// ---------------- problem constants (match reference) ----------------
constexpr int Bb  = 4;
constexpr int Hh  = 8;
constexpr int Tt  = 8192;
constexpr int Dd  = 64;
constexpr int NCc = 128;
constexpr int Wsz = 64;
constexpr int T2  = 2 * Tt;      // 16384 rows of x = concat(q,k)
constexpr int BH  = Bb * Hh;     // 32
constexpr float SCALE      = 0.125f;     // d^-0.5
constexpr float COMMIT     = 1.0e-4f;
constexpr float EPSV       = 1.0e-5f;
constexpr float NEG_BIG    = -3.0e38f;

typedef float v2f __attribute__((ext_vector_type(2)));
typedef float v8f __attribute__((ext_vector_type(8)));
typedef unsigned int u32x4 __attribute__((ext_vector_type(4)));
typedef unsigned int u32x8 __attribute__((ext_vector_type(8)));

// fp32 WMMA: D(16x16) = A(16x4) * B(4x16) + C   -> v_wmma_f32_16x16x4_f32
static __device__ __forceinline__ v8f wmma_f32(v2f a, v2f b, v8f c) {
    return __builtin_amdgcn_wmma_f32_16x16x4_f32(
        /*neg_a=*/false, a, /*neg_b=*/false, b,
        /*c_mod=*/(short)0, c, /*reuse_a=*/false, /*reuse_b=*/false);
}

// LDS byte offset of a __shared__ object (flat LDS addresses carry the
// offset in the low 32 bits; hardware adds LDS_BASE).
static __device__ __forceinline__ unsigned lds_off(const void* p) {
    return (unsigned)(unsigned long long)p;
}

// per-lane async copy: 16B  global -> LDS   (ASYNCcnt tracked)
static __device__ __forceinline__ void async_b128(unsigned lds, const void* gaddr) {
    asm volatile("global_load_async_to_lds_b128 %0, %1, off"
                 :: "v"(lds), "v"(gaddr) : "memory");
}
static __device__ __forceinline__ void wait_async0() {
    asm volatile("s_wait_asynccnt 0x0" ::: "memory");
}

// TDM tile load via descriptor groups in SGPRs (TENSORcnt tracked)
static __device__ __forceinline__ void tdm_load(u32x4 g0, u32x8 g1) {
    asm volatile("tensor_load_to_lds %0, %1" :: "s"(g0), "s"(g1) : "memory");
}

// build D# group0: count=1, lds addr, 57-bit global addr, type=2
static __device__ __forceinline__ u32x4 tdm_g0(unsigned lds, const void* gp) {
    unsigned long long ga = (unsigned long long)gp;
    u32x4 g0;
    g0.x = 1u;                                       // count=1, user mode
    g0.y = lds;                                      // lds_addr
    g0.z = (unsigned)ga;                             // global_addr[31:0]
    g0.w = ((unsigned)(ga >> 32) & 0x01FFFFFFu) | 0x80000000u; // addr[56:32] | type=2
    return g0;
}

// ---------------------------------------------------------------------
// K0: zero output accumulator, per-token counts, aux accumulator
// ---------------------------------------------------------------------
__global__ void __launch_bounds__(256)
k_init(float* __restrict__ out, float* __restrict__ counts, float* __restrict__ aux) {
    const size_t nOut = (size_t)BH * Tt * Dd + 1;
    const size_t nCnt = (size_t)BH * Tt;
    size_t stride = (size_t)gridDim.x * blockDim.x;
    for (size_t i = (size_t)blockIdx.x * blockDim.x + threadIdx.x; i < nOut; i += stride)
        out[i] = 0.0f;
    for (size_t i = (size_t)blockIdx.x * blockDim.x + threadIdx.x; i < nCnt; i += stride)
        counts[i] = 0.0f;
    if (blockIdx.x == 0 && threadIdx.x == 0) aux[0] = 0.0f;
}

// ---------------------------------------------------------------------
// K1: msq[h][c] = ||means[h,c,:]||^2
// ---------------------------------------------------------------------
__global__ void __launch_bounds__(256)
k_msq(const float* __restrict__ means, float* __restrict__ msq) {
    int i = blockIdx.x * blockDim.x + threadIdx.x;
    if (i < Hh * NCc) {
        const float* m = means + (size_t)i * Dd;
        float s = 0.0f;
        #pragma unroll 8
        for (int d = 0; d < Dd; ++d) { float x = m[d]; s += x * x; }
        msq[i] = s;
    }
}

// ---------------------------------------------------------------------
// K2: dists = l2norm(x) . means^T  (WMMA fp32), transposed store + argmax/aux
//   grid: (BH, T2/64), block: 128 threads = 4 waves; wave -> 16 rows x 128 cols
//   means tile staged by the TDM with the pad feature producing stride-65 rows
// ---------------------------------------------------------------------
__global__ void __launch_bounds__(128)
k_dists(const float* __restrict__ q, const float* __restrict__ kk_,
        const float* __restrict__ means, const float* __restrict__ msq,
        float* __restrict__ distsT, float* __restrict__ aux) {
    __shared__ float xs[64][65];   // normalized rows
    __shared__ float ms[128][65];  // means for this head (TDM-padded stride 65)

    const int bh   = blockIdx.x;
    const int h    = bh % Hh;
    const int row0 = blockIdx.y * 64;
    const int tid  = threadIdx.x;

    // --- TDM: means[h] (128x64 f32) -> ms with 1-dword pad every 64 dwords ---
    if (tid < 32) {
        u32x4 g0 = tdm_g0(lds_off(&ms[0][0]), means + (size_t)(h * NCc) * Dd);
        u32x8 g1;
        g1[0] = 0x00020000u      // data_size=4B
              | 0x00100000u      // pad_enable
              | 0x01400000u;     // pad_interval code 5 (=64 dwords), pad_amount 0 (=1 dword)
        g1[1] = 64u << 16;       // tensor_dim0 = 64
        g1[2] = 128u << 16;      // tensor_dim1 = 128
        g1[3] = 64u << 16;       // tile_dim0 = 64
        g1[4] = 128u;            // tile_dim1 = 128, tile_dim2 = 0
        g1[5] = 64u;             // tensor_dim0_stride = 64
        g1[6] = 0u;
        g1[7] = 0u;
        tdm_load(g0, g1);
    }

    // normalize 64 rows of x = concat(q,k) (two-pass to avoid scratch spill)
    if (tid < 64) {
        const int l = row0 + tid;
        const float* src = (l < Tt) ? (q   + ((size_t)bh * Tt + l)        * Dd)
                                    : (kk_ + ((size_t)bh * Tt + (l - Tt)) * Dd);
        __builtin_prefetch(src, 0, 0);
        float ss = 0.0f;
        #pragma unroll 8
        for (int d = 0; d < Dd; ++d) { float xv = src[d]; ss += xv * xv; }
        const float inv = 1.0f / fmaxf(sqrtf(ss), 1e-12f);
        #pragma unroll 8
        for (int d = 0; d < Dd; ++d) xs[tid][d] = src[d] * inv;
    }
    __builtin_amdgcn_s_wait_tensorcnt(0);
    __syncthreads();

    const int wave    = tid >> 5;
    const int lane    = tid & 31;
    const int laneLo  = lane & 15;
    const int hi      = lane >> 4;           // 0: lanes 0-15, 1: lanes 16-31
    const int rowBase = wave * 16;

    v8f acc[8] = {};                          // 16 rows x 128 cols accumulators
    #pragma unroll
    for (int kstep = 0; kstep < 16; ++kstep) {
        const int k0 = 4 * kstep + 2 * hi;    // A/B wave32 K split (ISA 7.12.2)
        v2f a;
        a.x = xs[rowBase + laneLo][k0];
        a.y = xs[rowBase + laneLo][k0 + 1];
        #pragma unroll
        for (int n = 0; n < 8; ++n) {
            v2f b;
            b.x = ms[16 * n + laneLo][k0];
            b.y = ms[16 * n + laneLo][k0 + 1];
            acc[n] = wmma_f32(a, b, acc[n]);
        }
    }

    // transposed store + per-row argmax -> commitment-loss contribution
    float auxLocal = 0.0f;
    #pragma unroll
    for (int j = 0; j < 8; ++j) {
        const int rowG = row0 + rowBase + j + 8 * hi;   // C layout: vgpr j = row j / j+8
        float bv = NEG_BIG; int bc = 0;
        #pragma unroll
        for (int n = 0; n < 8; ++n) {
            const float vv = acc[n][j];
            const int   c  = 16 * n + laneLo;
            distsT[((size_t)(bh * NCc + c)) * T2 + rowG] = vv;
            if (vv > bv || (vv == bv && c < bc)) { bv = vv; bc = c; }
        }
        #pragma unroll
        for (int m = 8; m >= 1; m >>= 1) {
            float ov = __shfl_xor(bv, m, 16);
            int   oc = __shfl_xor(bc, m, 16);
            if (ov > bv || (ov == bv && oc < bc)) { bv = ov; bc = oc; }
        }
        if (laneLo == 0) {
            // ||x - m||^2 = 1 + ||m||^2 - 2 x.m   (x is unit norm)
            auxLocal += 1.0f + msq[h * NCc + bc] - 2.0f * bv;
        }
    }
    if (laneLo == 0) atomicAdd(aux, auxLocal);
}

// ---------------------------------------------------------------------
// K3: top-64 row indices per (bh, cluster, side)   grid: (BH*NC, 2)
//   8192-float score row staged by a single TDM tile load
// ---------------------------------------------------------------------
__global__ void __launch_bounds__(256)
k_topk(const float* __restrict__ distsT, int* __restrict__ idxq, int* __restrict__ idxk) {
    __shared__ float vals[Tt];
    __shared__ float rv[256];
    __shared__ int   ri[256];

    const int bhc  = blockIdx.x;
    const int side = blockIdx.y;
    const int tid  = threadIdx.x;
    const float* src = distsT + (size_t)bhc * T2 + side * Tt;
    int* dst = (side ? idxk : idxq) + (size_t)bhc * Wsz;

    if (tid < 32) {
        u32x4 g0 = tdm_g0(lds_off(&vals[0]), src);
        u32x8 g1;
        g1[0] = 0x00020000u;       // data_size = 4B
        g1[1] = 8192u << 16;       // tensor_dim0 = 8192
        g1[2] = 1u << 16;          // tensor_dim1 = 1
        g1[3] = 8192u << 16;       // tile_dim0 = 8192
        g1[4] = 1u;                // tile_dim1 = 1
        g1[5] = 8192u;             // tensor_dim0_stride
        g1[6] = 0u;
        g1[7] = 0u;
        tdm_load(g0, g1);
    }
    __builtin_amdgcn_s_wait_tensorcnt(0);
    __syncthreads();

    for (int it = 0; it < Wsz; ++it) {
        float bv = NEG_BIG; int bi = 0x7fffffff;
        for (int i = tid; i < Tt; i += 256) {
            const float vv = vals[i];
            if (vv > bv || (vv == bv && i < bi)) { bv = vv; bi = i; }
        }
        rv[tid] = bv; ri[tid] = bi;
        __syncthreads();
        for (int s = 128; s > 0; s >>= 1) {
            if (tid < s) {
                if (rv[tid + s] > rv[tid] ||
                    (rv[tid + s] == rv[tid] && ri[tid + s] < ri[tid])) {
                    rv[tid] = rv[tid + s]; ri[tid] = ri[tid + s];
                }
            }
            __syncthreads();
        }
        if (tid == 0) { dst[it] = ri[0]; vals[ri[0]] = NEG_BIG; }
        __syncthreads();
    }
}

// ---------------------------------------------------------------------
// K4: clustered attention (async gather -> WMMA -> softmax -> WMMA -> scatter)
//   grid: BH*NC blocks, 128 threads = 4 waves; wave -> 16 query rows
//   data-dependent q/k/v gathers use per-lane global_load_async_to_lds_b128
//   LDS row stride 68 floats (272B) keeps B128 LDS destinations 16B aligned
// ---------------------------------------------------------------------
__global__ void __launch_bounds__(128)
k_attn(const float* __restrict__ q, const float* __restrict__ kk_,
       const float* __restrict__ v, const float* __restrict__ memk,
       const float* __restrict__ memv, const int* __restrict__ idxq,
       const int* __restrict__ idxk, float* __restrict__ num,
       float* __restrict__ counts) {
    __shared__ float qs[64][68];  // gathered q rows (unscaled; scale in softmax)
    __shared__ float ks[80][68];  // row 0 = mem_key, 1..64 gathered k, 65..79 = 0
    __shared__ float vs[80][68];  // same for v; ks region recycled for attn
    __shared__ int   qtok[64];

    const int bhc = blockIdx.x;
    const int bh  = bhc / NCc;
    const int c   = bhc % NCc;
    const int h   = bh % Hh;
    const int tid = threadIdx.x;

    const int* iq = idxq + (size_t)bhc * Wsz;
    const int* ik = idxk + (size_t)bhc * Wsz;

    if (tid < 64) {
        const int tk = iq[tid];
        qtok[tid] = tk;
        atomicAdd(&counts[(size_t)bh * Tt + tk], 1.0f);   // denominator scatter
        // memory slot row (NMEM == 1)
        ks[0][tid] = memk[(size_t)(h * NCc + c) * Dd + tid];
        vs[0][tid] = memv[(size_t)(h * NCc + c) * Dd + tid];
    }
    // zero the 15 padding rows (avoid 0*garbage = NaN in the second GEMM)
    for (int i = tid; i < 15 * 64; i += 128) {
        const int r = 65 + (i >> 6), d = i & 63;
        ks[r][d] = 0.0f;
        vs[r][d] = 0.0f;
    }
    // async gather: q rows (64 rows x 16 chunks of 16B)
    for (int cid = tid; cid < 64 * 16; cid += 128) {
        const int r = cid >> 4, c4 = (cid & 15) << 2;
        const int tk = iq[r];
        async_b128(lds_off(&qs[r][c4]),
                   q + ((size_t)bh * Tt + tk) * Dd + c4);
    }
    // async gather: k and v rows into rows 1..64
    for (int cid = tid; cid < 64 * 16; cid += 128) {
        const int r = cid >> 4, c4 = (cid & 15) << 2;
        const int tk = ik[r];
        async_b128(lds_off(&ks[r + 1][c4]),
                   kk_ + ((size_t)bh * Tt + tk) * Dd + c4);
        async_b128(lds_off(&vs[r + 1][c4]),
                   v + ((size_t)bh * Tt + tk) * Dd + c4);
    }
    wait_async0();
    __syncthreads();

    const int wave    = tid >> 5;
    const int lane    = tid & 31;
    const int laneLo  = lane & 15;
    const int hi      = lane >> 4;
    const int rowBase = wave * 16;

    // dots = qg . kg^T   : 16 rows x 80 cols (cols >= 65 are padding)
    v8f dacc[5] = {};
    #pragma unroll
    for (int kstep = 0; kstep < 16; ++kstep) {
        const int k0 = 4 * kstep + 2 * hi;
        v2f a;
        a.x = qs[rowBase + laneLo][k0];
        a.y = qs[rowBase + laneLo][k0 + 1];
        #pragma unroll
        for (int n = 0; n < 5; ++n) {
            v2f b;
            b.x = ks[16 * n + laneLo][k0];
            b.y = ks[16 * n + laneLo][k0 + 1];
            dacc[n] = wmma_f32(a, b, dacc[n]);
        }
    }
    __syncthreads();   // everyone done reading ks as B -> safe to recycle as attn

    float* attnW = &ks[0][0] + wave * 16 * 81;   // 16 x 80, stride 81 (fits in ks)

    #pragma unroll
    for (int j = 0; j < 8; ++j) {
        const int m = j + 8 * hi;   // local query row this accumulator slice owns
        float mx = NEG_BIG;
        #pragma unroll
        for (int n = 0; n < 5; ++n) {
            const int col = 16 * n + laneLo;
            const float dv = (col < 65) ? dacc[n][j] * SCALE : NEG_BIG;
            mx = fmaxf(mx, dv);
        }
        #pragma unroll
        for (int s = 8; s >= 1; s >>= 1) mx = fmaxf(mx, __shfl_xor(mx, s, 16));
        float se = 0.0f;
        #pragma unroll
        for (int n = 0; n < 5; ++n) {
            const int col = 16 * n + laneLo;
            const float e = (col < 65) ? __expf(dacc[n][j] * SCALE - mx) : 0.0f;
            dacc[n][j] = e;
            se += e;
        }
        #pragma unroll
        for (int s = 8; s >= 1; s >>= 1) se += __shfl_xor(se, s, 16);
        const float inv = 1.0f / se;
        #pragma unroll
        for (int n = 0; n < 5; ++n) {
            const int col = 16 * n + laneLo;
            attnW[m * 81 + col] = dacc[n][j] * inv;
        }
    }
    __syncthreads();

    // bo = attn(16x80) . vg(80x64)
    v8f oacc[4] = {};
    #pragma unroll
    for (int kstep = 0; kstep < 20; ++kstep) {
        const int k0 = 4 * kstep + 2 * hi;
        v2f a;
        a.x = attnW[laneLo * 81 + k0];
        a.y = attnW[laneLo * 81 + k0 + 1];
        #pragma unroll
        for (int n = 0; n < 4; ++n) {
            v2f b;
            b.x = vs[k0][16 * n + laneLo];
            b.y = vs[k0 + 1][16 * n + laneLo];
            oacc[n] = wmma_f32(a, b, oacc[n]);
        }
    }

    // numerator scatter (order-independent sum via float atomics)
    #pragma unroll
    for (int j = 0; j < 8; ++j) {
        const int rloc = rowBase + j + 8 * hi;
        const int tk   = qtok[rloc];
        float* dstRow = num + ((size_t)bh * Tt + tk) * Dd;
        #pragma unroll
        for (int n = 0; n < 4; ++n) {
            atomicAdd(dstRow + 16 * n + laneLo, oacc[n][j]);
        }
    }
}

// ---------------------------------------------------------------------
// K5: out = num / (den + eps); append aux loss scalar
// ---------------------------------------------------------------------
__global__ void __launch_bounds__(256)
k_final(float* __restrict__ out, const float* __restrict__ counts,
        const float* __restrict__ aux) {
    const size_t n = (size_t)BH * Tt * Dd;
    const size_t i = (size_t)blockIdx.x * blockDim.x + threadIdx.x;
    if (i < n) out[i] = out[i] / (counts[i >> 6] + EPSV);
    if (i == 0) {
        // mean over (b,h,2t,d) elements, times commitment weight
        out[n] = aux[0] * (COMMIT / (float)((size_t)BH * T2 * Dd));
    }
}

// ---------------------------------------------------------------------
extern "C" void kernel_launch(void* const* d_in, const int* in_sizes, int n_in,
                              void* d_out, int out_size, void* d_ws, size_t ws_size,
                              hipStream_t stream) {
    const float* q     = (const float*)d_in[0];
    const float* k     = (const float*)d_in[1];
    const float* v     = (const float*)d_in[2];
    const float* means = (const float*)d_in[3];
    const float* memk  = (const float*)d_in[4];
    const float* memv  = (const float*)d_in[5];
    float* out = (float*)d_out;

    // workspace layout (floats): distsT | msq | aux | counts | idxq | idxk
    float* distsT = (float*)d_ws;
    float* msq    = distsT + (size_t)BH * NCc * T2;      // 67,108,864 f
    float* aux    = msq + Hh * NCc;                      // 1,024 f
    float* counts = aux + 64;                            // padded
    int*   idxq   = (int*)(counts + (size_t)BH * Tt);    // 262,144 f
    int*   idxk   = idxq + (size_t)BH * NCc * Wsz;       // 262,144 i each

    k_init <<<4096, 256, 0, stream>>>(out, counts, aux);
    k_msq  <<<(Hh * NCc + 255) / 256, 256, 0, stream>>>(means, msq);
    k_dists<<<dim3(BH, T2 / 64), 128, 0, stream>>>(q, k, means, msq, distsT, aux);
    k_topk <<<dim3(BH * NCc, 2), 256, 0, stream>>>(distsT, idxq, idxk);
    k_attn <<<BH * NCc, 128, 0, stream>>>(q, k, v, memk, memv, idxq, idxk, out, counts);
    k_final<<<(int)(((size_t)BH * Tt * Dd + 255) / 256), 256, 0, stream>>>(out, counts, aux);
}